// GCN_936302870865
// MI455X (gfx1250) — compile-verified
//
#include <hip/hip_runtime.h>
#include <hip/hip_bf16.h>
#include <math.h>

typedef __attribute__((ext_vector_type(2))) float v2f;
typedef __attribute__((ext_vector_type(8))) float v8f;

#define TILES_PER_WAVE 4

// Hardware float atomic (global_atomic_add_f32), no CAS loop.
__device__ __forceinline__ void atomAddF(float* p, float v) {
    unsafeAtomicAdd(p, v);
}

// deg[i] = 1.0 (self-loop contribution)
__global__ void gcn_init_deg(float* __restrict__ deg, int n) {
    int i = blockIdx.x * blockDim.x + threadIdx.x;
    if (i < n) deg[i] = 1.0f;
}

// deg[col[e]] += 1 for the real edges
__global__ void gcn_count_deg(const int* __restrict__ col, float* __restrict__ deg, int E) {
    int e = blockIdx.x * blockDim.x + threadIdx.x;
    if (e < E) atomAddF(&deg[col[e]], 1.0f);
}

// dinv[i] = deg > 0 ? deg^-1/2 : 0   (in place)
__global__ void gcn_make_dinv(float* __restrict__ dinv, int n) {
    int i = blockIdx.x * blockDim.x + threadIdx.x;
    if (i < n) {
        float d = dinv[i];
        dinv[i] = (d > 0.0f) ? rsqrtf(d) : 0.0f;
    }
}

// hx[N,16] = x[N,3] @ W1[3,16] via V_WMMA_F32_16X16X4_F32.
// Each wave handles TILES_PER_WAVE consecutive 16-node tiles; B (= W1 padded to
// 4x16) is loaded once per wave and reused across all tiles (B is tile-invariant).
// A layout (16x4 f32): lanes 0-15 -> K=0,1 ; lanes 16-31 -> K=2,3 (2 VGPRs/lane).
// D layout (16x16 f32): lanes 0-15 -> M=0..7 in v[0..7]; lanes 16-31 -> M=8..15.
__global__ void gcn_gemm1_wmma(const float* __restrict__ x, const float* __restrict__ W1,
                               float* __restrict__ hx, int n_tiles, int n_nodes) {
    int wave  = (blockIdx.x * blockDim.x + threadIdx.x) >> 5;
    int lane  = threadIdx.x & 31;
    int tile0 = wave * TILES_PER_WAVE;
    if (tile0 >= n_tiles) return;         // wave-uniform: EXEC stays all-1s for WMMA

    int   half = lane >> 4;               // 0: K=0,1   1: K=2,3
    int   idx  = lane & 15;               // row (A) / col (B)
    int   k0   = half * 2;
    int   k1   = k0 + 1 - half;           // half=0 -> 1 ; half=1 -> 2 (valid addr)
    float pad  = half ? 0.0f : 1.0f;      // zero the K=3 pad lane branch-free

    // B operand: W1 padded to 4x16, invariant across tiles.
    v2f b;
    b.x = W1[k0 * 16 + idx];
    b.y = W1[k1 * 16 + idx] * pad;

#pragma unroll
    for (int t = 0; t < TILES_PER_WAVE; ++t) {
        int tile = tile0 + t;
        if (tile >= n_tiles) break;       // wave-uniform
        int m0 = tile * 16;

        int m = m0 + idx;
        if (m >= n_nodes) m = n_nodes - 1;  // clamp loads for a tail tile

        v2f a;
        a.x = x[(size_t)m * 3 + k0];
        a.y = x[(size_t)m * 3 + k1] * pad;

        v8f c = {};
        c = __builtin_amdgcn_wmma_f32_16x16x4_f32(
            /*neg_a=*/false, a, /*neg_b=*/false, b,
            /*c_mod=*/(short)0, c, /*reuse_a=*/false, /*reuse_b=*/false);

        float* dst = hx + (size_t)(m0 + half * 8) * 16 + idx;
        if (m0 + 16 <= n_nodes) {         // wave-uniform fast path: no per-store guards
#pragma unroll
            for (int r = 0; r < 8; ++r) dst[(size_t)r * 16] = c[r];
        } else {
#pragma unroll
            for (int r = 0; r < 8; ++r)
                if (m0 + half * 8 + r < n_nodes) dst[(size_t)r * 16] = c[r];
        }
    }
}

// Layer-1 propagate: out[col] += hx[row] * (dinv[row]*dinv[col]); e >= E are self-loops.
__global__ void gcn_scatter16(const int* __restrict__ row, const int* __restrict__ col,
                              const float* __restrict__ dinv, const float* __restrict__ hx,
                              float* __restrict__ out, int E, int N) {
    int e = blockIdx.x * blockDim.x + threadIdx.x;
    if (e >= E + N) return;
    int r, c;
    if (e < E) { r = row[e]; c = col[e]; } else { r = c = e - E; }
    float nrm = dinv[r] * dinv[c];
    const float4* src = (const float4*)(hx + (size_t)r * 16);
    float* dst = out + (size_t)c * 16;
#pragma unroll
    for (int q = 0; q < 4; ++q) {
        float4 v = src[q];
        atomAddF(dst + q * 4 + 0, v.x * nrm);
        atomAddF(dst + q * 4 + 1, v.y * nrm);
        atomAddF(dst + q * 4 + 2, v.z * nrm);
        atomAddF(dst + q * 4 + 3, v.w * nrm);
    }
}

// a1 = relu(a1 + b1)   (a1 is node-major [i*16+k])
__global__ void gcn_bias_relu(float* __restrict__ a, const float* __restrict__ b1, int total) {
    int i = blockIdx.x * blockDim.x + threadIdx.x;
    if (i < total) {
        float v = a[i] + b1[i & 15];
        a[i] = v > 0.0f ? v : 0.0f;
    }
}

// h2[N,2] = a1[N,16] @ W2[16,2]
__global__ void gcn_gemm2(const float* __restrict__ a1, const float* __restrict__ W2,
                          float* __restrict__ h2, int N) {
    int i = blockIdx.x * blockDim.x + threadIdx.x;
    if (i >= N) return;
    const float4* p = (const float4*)(a1 + (size_t)i * 16);
    float s0 = 0.0f, s1 = 0.0f;
#pragma unroll
    for (int q = 0; q < 4; ++q) {
        float4 v = p[q];
        s0 += v.x * W2[(q * 4 + 0) * 2 + 0] + v.y * W2[(q * 4 + 1) * 2 + 0]
            + v.z * W2[(q * 4 + 2) * 2 + 0] + v.w * W2[(q * 4 + 3) * 2 + 0];
        s1 += v.x * W2[(q * 4 + 0) * 2 + 1] + v.y * W2[(q * 4 + 1) * 2 + 1]
            + v.z * W2[(q * 4 + 2) * 2 + 1] + v.w * W2[(q * 4 + 3) * 2 + 1];
    }
    h2[(size_t)i * 2 + 0] = s0;
    h2[(size_t)i * 2 + 1] = s1;
}

// Layer-2 propagate (2 features per node)
__global__ void gcn_scatter2(const int* __restrict__ row, const int* __restrict__ col,
                             const float* __restrict__ dinv, const float* __restrict__ h2,
                             float* __restrict__ out2, int E, int N) {
    int e = blockIdx.x * blockDim.x + threadIdx.x;
    if (e >= E + N) return;
    int r, c;
    if (e < E) { r = row[e]; c = col[e]; } else { r = c = e - E; }
    float nrm = dinv[r] * dinv[c];
    float2 v = *(const float2*)(h2 + (size_t)r * 2);
    atomAddF(out2 + (size_t)c * 2 + 0, v.x * nrm);
    atomAddF(out2 + (size_t)c * 2 + 1, v.y * nrm);
}

// out = log_softmax(out2 + b2) over the 2 classes (numerically stable)
__global__ void gcn_logsoftmax(const float* __restrict__ out2, const float* __restrict__ b2,
                               float* __restrict__ out, int N) {
    int i = blockIdx.x * blockDim.x + threadIdx.x;
    if (i >= N) return;
    float v0 = out2[(size_t)i * 2 + 0] + b2[0];
    float v1 = out2[(size_t)i * 2 + 1] + b2[1];
    float m  = fmaxf(v0, v1);
    float lse = m + logf(expf(v0 - m) + expf(v1 - m));
    out[(size_t)i * 2 + 0] = v0 - lse;
    out[(size_t)i * 2 + 1] = v1 - lse;
}

extern "C" void kernel_launch(void* const* d_in, const int* in_sizes, int n_in,
                              void* d_out, int out_size, void* d_ws, size_t ws_size,
                              hipStream_t stream) {
    const float* x  = (const float*)d_in[0];   // [N,3]
    const int*   ei = (const int*)d_in[1];     // [2,E] int32
    const float* W1 = (const float*)d_in[2];   // [3,16]
    const float* b1 = (const float*)d_in[3];   // [16]
    const float* W2 = (const float*)d_in[4];   // [16,2]
    const float* b2 = (const float*)d_in[5];   // [2]

    const int N = in_sizes[0] / 3;
    const int E = in_sizes[1] / 2;
    const int* row = ei;
    const int* col = ei + E;

    // Workspace layout (floats): dinv[N] | hx[16N] | a1[16N]
    // h2[2N] and out2[2N] reuse the dead hx region after layer 1.
    float* ws   = (float*)d_ws;
    float* dinv = ws;
    float* hx   = ws + (size_t)N;
    float* a1   = ws + (size_t)N + 16 * (size_t)N;
    float* h2   = hx;                 // [2N], overwrites hx after it's consumed
    float* out2 = hx + 2 * (size_t)N; // [2N], also inside old hx region

    const int BLK = 256;
    const int n_tiles = (N + 15) / 16;

    // --- degrees / normalization ---
    gcn_init_deg<<<(N + BLK - 1) / BLK, BLK, 0, stream>>>(dinv, N);
    gcn_count_deg<<<(E + BLK - 1) / BLK, BLK, 0, stream>>>(col, dinv, E);
    gcn_make_dinv<<<(N + BLK - 1) / BLK, BLK, 0, stream>>>(dinv, N);

    // --- layer 1 ---
    hipMemsetAsync(a1, 0, 16 * (size_t)N * sizeof(float), stream);
    // 8 waves per 256-thread block, TILES_PER_WAVE 16-node WMMA tiles per wave
    int waves_per_blk  = BLK / 32;
    int tiles_per_blk  = waves_per_blk * TILES_PER_WAVE;
    gcn_gemm1_wmma<<<(n_tiles + tiles_per_blk - 1) / tiles_per_blk, BLK, 0, stream>>>(
        x, W1, hx, n_tiles, N);
    gcn_scatter16<<<(E + N + BLK - 1) / BLK, BLK, 0, stream>>>(row, col, dinv, hx, a1, E, N);
    gcn_bias_relu<<<(16 * N + BLK - 1) / BLK, BLK, 0, stream>>>(a1, b1, 16 * N);

    // --- layer 2 ---
    gcn_gemm2<<<(N + BLK - 1) / BLK, BLK, 0, stream>>>(a1, W2, h2, N);
    hipMemsetAsync(out2, 0, 2 * (size_t)N * sizeof(float), stream);
    gcn_scatter2<<<(E + N + BLK - 1) / BLK, BLK, 0, stream>>>(row, col, dinv, h2, out2, E, N);

    // --- log_softmax -> d_out ---
    gcn_logsoftmax<<<(N + BLK - 1) / BLK, BLK, 0, stream>>>(out2, b2, (float*)d_out, N);
}